// MemNet_41128606827068
// MI455X (gfx1250) — compile-verified
//
#include <hip/hip_runtime.h>
#include <hip/hip_bf16.h>
#include <math.h>

typedef __attribute__((ext_vector_type(2))) float v2f;
typedef __attribute__((ext_vector_type(8))) float v8f;

#define B_   16
#define M_   512
#define S_   32
#define E_   128
#define V_   32000
#define NILIDX (V_ - 1)

// enc[s,e] = 1 + 4*((e+1 - E/2)*(s+1 - S/2))/E/S
__device__ __forceinline__ float enc_val(int s, int e) {
    float a = (float)(e + 1) - 64.0f;   // E/2
    float b = (float)(s + 1) - 16.0f;   // S/2
    return 1.0f + 4.0f * (a * b) / 128.0f / 32.0f;
}

// One block per (b,m); thread = channel e. Gathers both tables with indices
// staged in LDS; rows are contiguous 512B -> coalesced across the 4 waves.
__global__ void embed_kernel(const int* __restrict__ stories,
                             const float* __restrict__ st_tab,   // [V-1, E]
                             const float* __restrict__ o_tab,    // [V-1, E]
                             const float* __restrict__ mem_bias, // [M, E]
                             float* __restrict__ memory,         // [B*M, E]
                             float* __restrict__ output)         // [B*M, E]
{
    const int bm = blockIdx.x;          // 0 .. B*M-1
    const int m  = bm & (M_ - 1);
    const int e  = threadIdx.x;         // 0 .. 127
    __shared__ int idx[S_];
    if (threadIdx.x < S_) idx[threadIdx.x] = stories[bm * S_ + threadIdx.x];
    __syncthreads();
    float am = 0.0f, ao = 0.0f;
    #pragma unroll
    for (int s = 0; s < S_; ++s) {
        int ix = idx[s];
        if (ix < NILIDX) {              // index V-1 is the nil (zero) row
            float w = enc_val(s, e);
            am = fmaf(st_tab[(size_t)ix * E_ + e], w, am);
            ao = fmaf(o_tab [(size_t)ix * E_ + e], w, ao);
        }
    }
    memory[(size_t)bm * E_ + e] = am + mem_bias[(size_t)m * E_ + e];
    output[(size_t)bm * E_ + e] = ao;
}

__global__ void qembed_kernel(const int* __restrict__ queries,
                              const float* __restrict__ q_tab,
                              float* __restrict__ qbuf)           // [B, E]
{
    const int b = blockIdx.x;
    const int e = threadIdx.x;
    __shared__ int idx[S_];
    if (threadIdx.x < S_) idx[threadIdx.x] = queries[b * S_ + threadIdx.x];
    __syncthreads();
    float acc = 0.0f;
    #pragma unroll
    for (int s = 0; s < S_; ++s) {
        int ix = idx[s];
        if (ix < NILIDX) acc = fmaf(q_tab[(size_t)ix * E_ + e], enc_val(s, e), acc);
    }
    qbuf[b * E_ + e] = acc;
}

// One block per batch: scores = memory . q, softmax over M, layer_out =
// output^T probs, u = q + layer_out  (stored for the WMMA projection).
__global__ void hop_probs_kernel(const float* __restrict__ memory,
                                 const float* __restrict__ output,
                                 const float* __restrict__ qbuf,
                                 float* __restrict__ ubuf)        // [B, E]
{
    const int b = blockIdx.x;
    const int t = threadIdx.x;          // 0..255
    const int lane = t & 31, wave = t >> 5;
    __shared__ float qs[E_];
    __shared__ float sc[M_];
    __shared__ float red[256];
    __shared__ float lo[2][E_];

    if (t < E_) qs[t] = qbuf[b * E_ + t];
    __syncthreads();

    // scores: wave-per-row, lane-parallel dot over E (coalesced row reads)
    const float* memb = memory + (size_t)b * M_ * E_;
    for (int m = wave * 64; m < wave * 64 + 64; ++m) {
        const float* row = memb + (size_t)m * E_;
        float p = row[lane]      * qs[lane]
                + row[lane + 32] * qs[lane + 32]
                + row[lane + 64] * qs[lane + 64]
                + row[lane + 96] * qs[lane + 96];
        #pragma unroll
        for (int off = 16; off > 0; off >>= 1) p += __shfl_xor(p, off, 32);
        if (lane == 0) sc[m] = p;
    }
    __syncthreads();

    // softmax: max
    red[t] = fmaxf(sc[t], sc[t + 256]);
    __syncthreads();
    for (int s = 128; s > 0; s >>= 1) {
        if (t < s) red[t] = fmaxf(red[t], red[t + s]);
        __syncthreads();
    }
    float mx = red[0];
    __syncthreads();
    // exp + sum
    float e0 = expf(sc[t] - mx), e1 = expf(sc[t + 256] - mx);
    sc[t] = e0; sc[t + 256] = e1;
    red[t] = e0 + e1;
    __syncthreads();
    for (int s = 128; s > 0; s >>= 1) {
        if (t < s) red[t] += red[t + s];
        __syncthreads();
    }
    float inv = 1.0f / red[0];

    // layer_out: thread owns channel e, split M across two halves
    const int e = t & (E_ - 1), half = t >> 7;
    const float* outb = output + (size_t)b * M_ * E_;
    float acc = 0.0f;
    for (int m = half * 256; m < half * 256 + 256; ++m)
        acc = fmaf(outb[(size_t)m * E_ + e], sc[m], acc);
    lo[half][e] = acc;
    __syncthreads();
    if (t < E_) {
        float u = qs[t] + (lo[0][t] + lo[1][t]) * inv;
        ubuf[b * E_ + t] = u;
    }
}

// q_next = U[16,128] @ W[128,128] via V_WMMA_F32_16X16X4_F32.
// 8 waves, each owns a 16-column tile; 32 K-steps of depth 4.
__global__ void hop_gemm_kernel(const float* __restrict__ ubuf,
                                const float* __restrict__ W,
                                float* __restrict__ qbuf)
{
    __shared__ float U[B_ * E_];
    const int t = threadIdx.x;
    for (int i = t; i < B_ * E_; i += 256) U[i] = ubuf[i];
    __syncthreads();
    const int wave = t >> 5, lane = t & 31;
    const int n0 = wave * 16;
    const int r = lane & 15, col = lane & 15, kb = (lane >> 4) * 2;
    v8f c = {};
    #pragma unroll
    for (int kk = 0; kk < 32; ++kk) {
        int k = kk * 4 + kb;
        v2f a, bb;
        a.x  = U[r * E_ + k];
        a.y  = U[r * E_ + k + 1];
        bb.x = W[(size_t)k * E_ + n0 + col];
        bb.y = W[(size_t)(k + 1) * E_ + n0 + col];
        c = __builtin_amdgcn_wmma_f32_16x16x4_f32(false, a, false, bb,
                                                  (short)0, c, false, false);
    }
    const int rowbase = (lane >> 4) * 8;
    #pragma unroll
    for (int v = 0; v < 8; ++v)
        qbuf[(rowbase + v) * E_ + n0 + col] = c[v];
}

// out = relu(q) @ w_final[128, 32000]; relu folded into the LDS stage of A.
// 2000 16-col tiles -> 250 blocks x 8 waves; coalesced w_final stripes.
__global__ void final_gemm_kernel(const float* __restrict__ qbuf,
                                  const float* __restrict__ Wf,
                                  float* __restrict__ out)
{
    __shared__ float A[B_ * E_];
    const int t = threadIdx.x;
    for (int i = t; i < B_ * E_; i += 256) A[i] = fmaxf(qbuf[i], 0.0f);
    __syncthreads();
    const int wave = t >> 5, lane = t & 31;
    const int tile = blockIdx.x * 8 + wave;     // 0..1999
    const int n0 = tile * 16;
    const int r = lane & 15, col = lane & 15, kb = (lane >> 4) * 2;
    v8f c = {};
    #pragma unroll
    for (int kk = 0; kk < 32; ++kk) {
        int k = kk * 4 + kb;
        v2f a, bb;
        a.x  = A[r * E_ + k];
        a.y  = A[r * E_ + k + 1];
        bb.x = Wf[(size_t)k * V_ + n0 + col];
        bb.y = Wf[(size_t)(k + 1) * V_ + n0 + col];
        c = __builtin_amdgcn_wmma_f32_16x16x4_f32(false, a, false, bb,
                                                  (short)0, c, false, false);
    }
    const int rowbase = (lane >> 4) * 8;
    #pragma unroll
    for (int v = 0; v < 8; ++v)
        out[(size_t)(rowbase + v) * V_ + n0 + col] = c[v];
}

extern "C" void kernel_launch(void* const* d_in, const int* in_sizes, int n_in,
                              void* d_out, int out_size, void* d_ws, size_t ws_size,
                              hipStream_t stream) {
    const int*   queries  = (const int*)d_in[0];
    const int*   stories  = (const int*)d_in[1];
    const float* q_tab    = (const float*)d_in[2];
    const float* st_tab   = (const float*)d_in[3];
    const float* mem_bias = (const float*)d_in[4];
    const float* o_tab    = (const float*)d_in[5];
    const float* w_int    = (const float*)d_in[6];
    const float* w_out    = (const float*)d_in[7];
    const float* w_fin    = (const float*)d_in[8];
    float* out = (float*)d_out;

    float* ws     = (float*)d_ws;
    float* memory = ws;                                // B*M*E = 1048576 f32
    float* output = ws + (size_t)B_ * M_ * E_;         // 1048576 f32
    float* qbuf   = output + (size_t)B_ * M_ * E_;     // 2048 f32
    float* ubuf   = qbuf + B_ * E_;                    // 2048 f32

    embed_kernel<<<B_ * M_, E_, 0, stream>>>(stories, st_tab, o_tab, mem_bias,
                                             memory, output);
    qembed_kernel<<<B_, E_, 0, stream>>>(queries, q_tab, qbuf);

    for (int hop = 0; hop < 3; ++hop) {
        hop_probs_kernel<<<B_, 256, 0, stream>>>(memory, output, qbuf, ubuf);
        const float* W = (hop == 2) ? w_out : w_int;
        hop_gemm_kernel<<<1, 256, 0, stream>>>(ubuf, W, qbuf);
    }

    final_gemm_kernel<<<V_ / 16 / 8, 256, 0, stream>>>(qbuf, w_fin, out);
}